// GWDLoss_14130442404018
// MI455X (gfx1250) — compile-verified
//
#include <hip/hip_runtime.h>
#include <hip/hip_bf16.h>
#include <math.h>

#define NBLOCKS  1024
#define NTHREADS 256
#define NWAVES   (NTHREADS / 32)   // wave32 on gfx1250

typedef __attribute__((ext_vector_type(2))) float v2f;
typedef __attribute__((ext_vector_type(8))) float v8f;

// Per-row GWD loss (FUN='log', TAU=1, ALPHA=1).
// Double-angle identity: tr(Sp*St) = 0.5*[ (ap+bp)(at+bt) +
// (ap-bp)(at-bt)*cos(2(rp-rt)) ]  -> one cosine per row, no sincos.
// All transcendentals are single-instruction TRANS ops (v_cos/v_sqrt/v_rsq/
// v_log/v_rcp), ~1-2 ulp each; errors wash out in the 2M-term mean.
__device__ __forceinline__ float row_loss(const float* __restrict__ p,
                                          const float* __restrict__ t) {
    const float xp = p[0], yp = p[1], wp = p[2], hp = p[3], rp = p[4];
    const float xt = t[0], yt = t[1], wt = t[2], ht = t[3], rt = t[4];

    const float dx = xp - xt, dy = yp - yt;
    const float xy = dx * dx + dy * dy;

    const float hwp = 0.5f * wp, hhp = 0.5f * hp;
    const float ap  = hwp * hwp, bp  = hhp * hhp;     // eigvals of Sigma_p
    const float hwt = 0.5f * wt, hht = 0.5f * ht;
    const float at  = hwt * hwt, bt  = hht * hht;     // eigvals of Sigma_t

    const float sum_p = ap + bp, dif_p = ap - bp;
    const float sum_t = at + bt, dif_t = at - bt;

    // tr(Sp^1/2 St Sp^1/2) = tr(Sp St): one cos of the angle difference.
    const float c2 = __cosf(2.0f * (rp - rt));        // arg in (-2,2)
    const float tr_cross = 0.5f * (sum_p * sum_t + dif_p * dif_t * c2);

    // det(cross) = (hwp*hhp*hwt*hht)^2 -> sqrt(clip(det)) = positive product
    const float sq_det = (hwp * hhp) * (hwt * hht);

    const float term = __builtin_amdgcn_sqrtf(fmaxf(tr_cross + 2.0f * sq_det, 0.0f));
    const float whr  = (sum_p + sum_t) - 2.0f * term; // traces rotation-invariant
    float dist = __builtin_amdgcn_sqrtf(fmaxf(xy + whr, 0.0f));

    // scale = (wp*hp*wt*ht)^(1/4);  1/scale = v_rsq(v_sqrt(prod))
    const float prod      = (wp * hp) * (wt * ht);
    const float inv_scale = __builtin_amdgcn_rsqf(__builtin_amdgcn_sqrtf(prod));
    dist = dist * inv_scale;

    const float L = __logf(1.0f + dist);              // FUN == 'log' (log1p)
    return 1.0f - __builtin_amdgcn_rcpf(1.0f + L);    // TAU == 1
}

// Wave32 reduction via v_wmma_f32_16x16x4_f32:
// A: lane m holds acc in VGPR0 (VGPR1 = 0); B = ones.
// D[m,n] = acc_m + acc_{m+16} exactly (other A terms are 0; f32 RNE).
// Each lane sums its 8 D regs (rows 0-7 / 8-15), xor-16 shuffle-add finishes.
__device__ __forceinline__ float wave_reduce_wmma(float acc) {
    v2f a; a.x = acc;  a.y = 0.0f;
    v2f b; b.x = 1.0f; b.y = 1.0f;
    v8f c = {};
    c = __builtin_amdgcn_wmma_f32_16x16x4_f32(
            /*neg_a=*/false, a, /*neg_b=*/false, b,
            /*c_mod=*/(short)0, c, /*reuse_a=*/false, /*reuse_b=*/false);
    float s = ((c[0] + c[1]) + (c[2] + c[3])) + ((c[4] + c[5]) + (c[6] + c[7]));
    s += __shfl_xor(s, 16, 32);
    return s;
}

__device__ __forceinline__ float block_reduce(float acc, float* smem) {
    const int lane = threadIdx.x & 31;
    const int wid  = threadIdx.x >> 5;
    const float ws = wave_reduce_wmma(acc);        // full EXEC here (no divergence)
    if (lane == 0) smem[wid] = ws;
    __syncthreads();
    float v = 0.0f;
    if (wid == 0) {
        v = (lane < NWAVES) ? smem[lane] : 0.0f;
        #pragma unroll
        for (int off = NWAVES >> 1; off > 0; off >>= 1)
            v += __shfl_xor(v, off, 32);
    }
    return v;                                      // valid on thread 0
}

__global__ void gwd_partial_kernel(const float* __restrict__ pred,
                                   const float* __restrict__ target,
                                   const float* __restrict__ weight,
                                   float* __restrict__ partial, int n) {
    const int tid    = blockIdx.x * blockDim.x + threadIdx.x;
    const int stride = gridDim.x * blockDim.x;
    const int groups = n >> 2;                     // 4 rows per group (80B, 16B-aligned)

    float acc = 0.0f;
    for (int g = tid; g < groups; g += stride) {
        const float4* p4 = reinterpret_cast<const float4*>(pred   + (size_t)g * 20);
        const float4* t4 = reinterpret_cast<const float4*>(target + (size_t)g * 20);
        const float4  wv = *reinterpret_cast<const float4*>(weight + (size_t)g * 4);

        alignas(16) float pb[20];
        alignas(16) float tb[20];
        #pragma unroll
        for (int k = 0; k < 5; ++k) {
            reinterpret_cast<float4*>(pb)[k] = p4[k];   // global_load_b128 x5
            reinterpret_cast<float4*>(tb)[k] = t4[k];   // global_load_b128 x5
        }

        // keep the load pipe deep for the next grid-stride pass
        const int gn = g + stride;
        if (gn < groups) {
            __builtin_prefetch(pred   + (size_t)gn * 20, 0, 1);
            __builtin_prefetch(target + (size_t)gn * 20, 0, 1);
        }

        acc += row_loss(pb +  0, tb +  0) * wv.x;
        acc += row_loss(pb +  5, tb +  5) * wv.y;
        acc += row_loss(pb + 10, tb + 10) * wv.z;
        acc += row_loss(pb + 15, tb + 15) * wv.w;
    }

    // tail rows (n % 4) — generic safety; N=2M has none
    if (tid == 0) {
        for (int i = groups * 4; i < n; ++i) {
            float pb[5], tb[5];
            #pragma unroll
            for (int k = 0; k < 5; ++k) {
                pb[k] = pred[(size_t)i * 5 + k];
                tb[k] = target[(size_t)i * 5 + k];
            }
            acc += row_loss(pb, tb) * weight[i];
        }
    }

    __shared__ float smem[NWAVES];
    const float v = block_reduce(acc, smem);
    if (threadIdx.x == 0) partial[blockIdx.x] = v;
}

__global__ void gwd_final_kernel(const float* __restrict__ partial,
                                 float* __restrict__ out,
                                 int nparts, float inv_n) {
    float acc = 0.0f;
    for (int i = threadIdx.x; i < nparts; i += blockDim.x)
        acc += partial[i];
    __shared__ float smem[NWAVES];
    const float v = block_reduce(acc, smem);
    if (threadIdx.x == 0) out[0] = v * inv_n;      // LOSS_WEIGHT == 1
}

extern "C" void kernel_launch(void* const* d_in, const int* in_sizes, int n_in,
                              void* d_out, int out_size, void* d_ws, size_t ws_size,
                              hipStream_t stream) {
    const float* pred   = (const float*)d_in[0];
    const float* target = (const float*)d_in[1];
    const float* weight = (const float*)d_in[2];
    const int n = in_sizes[2];                     // N rows (weight is (N,))

    float* partial = (float*)d_ws;                 // NBLOCKS floats of scratch

    gwd_partial_kernel<<<NBLOCKS, NTHREADS, 0, stream>>>(pred, target, weight, partial, n);
    gwd_final_kernel<<<1, NTHREADS, 0, stream>>>(partial, (float*)d_out, NBLOCKS,
                                                 1.0f / (float)n);
}